// SEST_28080496181753
// MI455X (gfx1250) — compile-verified
//
#include <hip/hip_runtime.h>
#include <math.h>

typedef __attribute__((ext_vector_type(2))) float v2f;
typedef __attribute__((ext_vector_type(8))) float v8f;

#define B_TILE   16
#define FDIM     39
#define EDIM     64
#define RDIM     13
#define ROWS     (B_TILE * FDIM)   // 624 rows per block
#define NTHREADS 640               // 20 waves; 624 active data rows
#define KP1      40                // GEMM1 K pad (39 -> 40, 10 wmma steps)
#define NP1      16                // GEMM1 N pad (13 -> 16)
#define KP2      16                // GEMM2 K pad (13 -> 16, 4 wmma steps)
#define NP2      48                // GEMM2 N pad (39 -> 48, 3 n-tiles)

__global__ __launch_bounds__(NTHREADS) void sest_fused_kernel(
    const float* __restrict__ in,
    const float* __restrict__ W1, const float* __restrict__ W2,
    const float* __restrict__ W3, const float* __restrict__ W4,
    float* __restrict__ out)
{
    __shared__ float Zs [B_TILE][KP1];      // mean,      zero-padded K
    __shared__ float AMs[B_TILE][KP1];      // abs-mean,  zero-padded K
    __shared__ float W1s[KP1][NP1];         // zero-padded 39x13 -> 40x16
    __shared__ float W3s[KP1][NP1];
    __shared__ float W2s[KP2][NP2];         // zero-padded 13x39 -> 16x48
    __shared__ float W4s[KP2][NP2];
    __shared__ float Hs [2][B_TILE][NP1];   // hidden activations per branch
    __shared__ float A2s[B_TILE][FDIM];     // gate
    __shared__ float THs[B_TILE][FDIM];     // threshold = abs_mean * A2_

    const int t = threadIdx.x;

    // ---- stage weights (zero-padded) + zero Z/AM pads ----
    // 16*40 == 640 == NTHREADS, KP1*NP1 == 640
    Zs [t / KP1][t % KP1] = 0.0f;
    AMs[t / KP1][t % KP1] = 0.0f;
    {
        const int k = t / NP1, n = t % NP1;
        const bool ok = (k < FDIM) && (n < RDIM);
        W1s[k][n] = ok ? W1[k * RDIM + n] : 0.0f;
        W3s[k][n] = ok ? W3[k * RDIM + n] : 0.0f;
    }
    for (int idx = t; idx < KP2 * NP2; idx += NTHREADS) {
        const int k = idx / NP2, n = idx % NP2;
        const bool ok = (k < RDIM) && (n < FDIM);
        W2s[k][n] = ok ? W2[k * FDIM + n] : 0.0f;
        W4s[k][n] = ok ? W4[k * FDIM + n] : 0.0f;
    }

    // ---- phase 1: one (b,f) row of 64 floats per thread, kept in VGPRs ----
    float4 x[16];
    float s = 0.0f, sa = 0.0f;
    const int    r       = t;                                    // row in tile
    const size_t rowbase = ((size_t)blockIdx.x * ROWS + r) * EDIM;
    if (r < ROWS) {
        const float4* p = (const float4*)(in + rowbase);
#pragma unroll
        for (int i = 0; i < 16; ++i) {
            x[i] = p[i];
            s  += x[i].x + x[i].y + x[i].z + x[i].w;
            sa += fabsf(x[i].x) + fabsf(x[i].y) + fabsf(x[i].z) + fabsf(x[i].w);
        }
    }
    __syncthreads();   // zero-init of Zs/AMs complete before data writes
    if (r < ROWS) {
        const int bl = r / FDIM, f = r % FDIM;
        Zs [bl][f] = s  * (1.0f / EDIM);
        AMs[bl][f] = sa * (1.0f / EDIM);
    }
    __syncthreads();

    const int wid  = t >> 5;
    const int lane = t & 31;
    const int m    = lane & 15;     // row (A) or col (B/C) index
    const int kh   = lane >> 4;     // K half-select for A/B fragments

    // ---- GEMM1: [16x40] @ [40x16], wave0 = gating branch, wave1 = abs branch ----
    if (wid < 2) {
        const float (*Xs)[KP1] = (wid == 0) ? Zs  : AMs;
        const float (*Ws)[NP1] = (wid == 0) ? W1s : W3s;
        v8f acc = {};
#pragma unroll
        for (int k0 = 0; k0 < KP1 / 4; ++k0) {
            const int kb = 4 * k0 + 2 * kh;
            v2f a; a[0] = Xs[m][kb];     a[1] = Xs[m][kb + 1];
            v2f b; b[0] = Ws[kb][m];     b[1] = Ws[kb + 1][m];
            acc = __builtin_amdgcn_wmma_f32_16x16x4_f32(
                false, a, false, b, (short)0, acc, false, false);
        }
#pragma unroll
        for (int v = 0; v < 8; ++v)
            Hs[wid][v + 8 * kh][m] = fmaxf(acc[v], 0.0f);   // relu
    }
    __syncthreads();

    // ---- GEMM2: [16x16] @ [16x48] in 3 n-tiles ----
    if (wid < 2) {
        const float (*Ws)[NP2] = (wid == 0) ? W2s : W4s;
#pragma unroll
        for (int nt = 0; nt < 3; ++nt) {
            v8f acc = {};
#pragma unroll
            for (int k0 = 0; k0 < KP2 / 4; ++k0) {
                const int kb = 4 * k0 + 2 * kh;
                v2f a; a[0] = Hs[wid][m][kb];     a[1] = Hs[wid][m][kb + 1];
                v2f b; b[0] = Ws[kb][nt * 16 + m]; b[1] = Ws[kb + 1][nt * 16 + m];
                acc = __builtin_amdgcn_wmma_f32_16x16x4_f32(
                    false, a, false, b, (short)0, acc, false, false);
            }
            const int col = nt * 16 + m;
            if (col < FDIM) {
#pragma unroll
                for (int v = 0; v < 8; ++v) {
                    const int   row = v + 8 * kh;
                    const float val = fmaxf(acc[v], 0.0f);  // relu
                    if (wid == 0) A2s[row][col] = val;
                    else          THs[row][col] = AMs[row][col] * val;
                }
            }
        }
    }
    __syncthreads();

    // ---- phase 3: elementwise from registers, stream V | R out ----
    if (r < ROWS) {
        const int   bl = r / FDIM, f = r % FDIM;
        const float a2 = A2s[bl][f];
        const float th = THs[bl][f];
        float4* po = (float4*)(out + rowbase * 2);   // output row = 128 floats
#pragma unroll
        for (int i = 0; i < 16; ++i) {
            const float4 xv = x[i];
            float4 Vv, Rv;
            Vv.x = xv.x * a2; Vv.y = xv.y * a2; Vv.z = xv.z * a2; Vv.w = xv.w * a2;
            Rv.x = copysignf(fmaxf(fabsf(xv.x) - th, 0.0f), xv.x);
            Rv.y = copysignf(fmaxf(fabsf(xv.y) - th, 0.0f), xv.y);
            Rv.z = copysignf(fmaxf(fabsf(xv.z) - th, 0.0f), xv.z);
            Rv.w = copysignf(fmaxf(fabsf(xv.w) - th, 0.0f), xv.w);
            po[i]      = Vv;   // V half  [0,64)
            po[16 + i] = Rv;   // R half  [64,128)
        }
    }
}

extern "C" void kernel_launch(void* const* d_in, const int* in_sizes, int n_in,
                              void* d_out, int out_size, void* d_ws, size_t ws_size,
                              hipStream_t stream) {
    (void)n_in; (void)out_size; (void)d_ws; (void)ws_size;
    const float* in = (const float*)d_in[0];
    const float* W1 = (const float*)d_in[1];
    const float* W2 = (const float*)d_in[2];
    const float* W3 = (const float*)d_in[3];
    const float* W4 = (const float*)d_in[4];
    float* out = (float*)d_out;

    const int nblocks = in_sizes[0] / (ROWS * EDIM);   // 16384*39*64 / 39936 = 1024
    sest_fused_kernel<<<nblocks, NTHREADS, 0, stream>>>(in, W1, W2, W3, W4, out);
}